// ScaledDotProductAttention_11776800325805
// MI455X (gfx1250) — compile-verified
//
#include <hip/hip_runtime.h>

typedef float v2f __attribute__((ext_vector_type(2)));
typedef float v8f __attribute__((ext_vector_type(8)));

#define DEV_INLINE __device__ __forceinline__

namespace {
constexpr int Bn = 2;        // batch
constexpr int Gn = 32;       // groups
constexpr int HD = 128;      // hidden per group
constexpr int Sn = 4096;     // sequence length
constexpr int NH = 8;        // heads
constexpr int Dn = 16;       // d_k = HD / NH
constexpr int SC = 8;        // s-positions per block == waves per block
constexpr int RQ = SC * Dn + 2;  // 130 words: row stride for sQ/sK  [g][s][d]+pad
constexpr int RA = Gn + 2;       // 34 words: row stride for sA [g][f]+pad and sV [s*16+d][f]+pad
}

DEV_INLINE v8f wmma4(v2f a, v2f b, v8f c) {
  // V_WMMA_F32_16X16X4_F32 : D = A(16x4) * B(4x16) + C(16x16)
  return __builtin_amdgcn_wmma_f32_16x16x4_f32(false, a, false, b, (short)0, c,
                                               false, false);
}

DEV_INLINE v2f lds2(const float* p) { return *(const v2f*)p; }

DEV_INLINE float red_max16(float x) {
  x = fmaxf(x, __shfl_xor(x, 1, 32));
  x = fmaxf(x, __shfl_xor(x, 2, 32));
  x = fmaxf(x, __shfl_xor(x, 4, 32));
  x = fmaxf(x, __shfl_xor(x, 8, 32));
  return x;
}
DEV_INLINE float red_sum16(float x) {
  x += __shfl_xor(x, 1, 32);
  x += __shfl_xor(x, 2, 32);
  x += __shfl_xor(x, 4, 32);
  x += __shfl_xor(x, 8, 32);
  return x;
}

__global__ __launch_bounds__(256)
void gqa_groupattn_kernel(const float* __restrict__ q,
                          const float* __restrict__ k,
                          const float* __restrict__ v,
                          float* __restrict__ out) {
  __shared__ __align__(16) float sQ[Gn * RQ];       // [g][s][d]; reused as X stage [g*16+d][s]
  __shared__ __align__(16) float sK[Gn * RQ];       // [g][s][d]
  __shared__ __align__(16) float sV[SC * Dn * RA];  // [s*16+d][f]
  __shared__ __align__(16) float sA[SC * Gn * RA];  // per-wave attn [g][f]; reused as mean stage

  const int tid  = threadIdx.x;
  const int wave = tid >> 5;    // == local s index
  const int lane = tid & 31;
  const int lm   = lane & 15;
  const int lh   = lane >> 4;   // lane half: 0 or 1
  const int b    = blockIdx.y;
  const int s0   = blockIdx.x * SC;

  float* outx = out;                                   // (B, 4096, S)
  float* outa = out + (size_t)Bn * Gn * HD * Sn;       // (B, 32, 32, S)
  float* outl = outa + (size_t)Bn * Gn * Gn * Sn;      // (B, 32, 32, S)

  v8f ml[2][2], ma[2][2];
  #pragma unroll
  for (int r = 0; r < 2; ++r)
    #pragma unroll
    for (int c = 0; c < 2; ++c) {
      ml[r][c] = (v8f){0, 0, 0, 0, 0, 0, 0, 0};
      ma[r][c] = (v8f){0, 0, 0, 0, 0, 0, 0, 0};
    }

  for (int h = 0; h < NH; ++h) {
    __syncthreads();  // previous iteration's LDS consumers done before restaging
    // ---- stage Q (pre-scaled), K, V : coalesced over s (innermost in global) ----
    {
      const int g  = tid >> 3;  // 0..31
      const int si = tid & 7;   // 0..7
      const size_t base = ((size_t)(b * Gn + g) * HD + h * Dn) * Sn + s0 + si;
      #pragma unroll
      for (int d = 0; d < Dn; ++d) {
        const size_t gi = base + (size_t)d * Sn;
        sQ[g * RQ + si * Dn + d]   = q[gi] * 0.25f;   // fold scale = d_k^-0.5
        sK[g * RQ + si * Dn + d]   = k[gi];
        sV[(si * Dn + d) * RA + g] = v[gi];
      }
    }
    __syncthreads();

    // ---- logits: 32x32 = 2x2 tiles of 16x16, K=16 via 4 chained wmma ----
    v2f aq[2][4], bk[2][4];
    #pragma unroll
    for (int r = 0; r < 2; ++r)
      #pragma unroll
      for (int c = 0; c < 4; ++c) {
        // A frag: [M = r*16+lm][K = c*4 + 2*lh + j]; B frag identical gather (B = K^T)
        aq[r][c] = lds2(&sQ[(r * 16 + lm) * RQ + wave * Dn + c * 4 + 2 * lh]);
        bk[r][c] = lds2(&sK[(r * 16 + lm) * RQ + wave * Dn + c * 4 + 2 * lh]);
      }
    v8f L[2][2];
    #pragma unroll
    for (int r = 0; r < 2; ++r)
      #pragma unroll
      for (int ct = 0; ct < 2; ++ct) {
        v8f acc = {0, 0, 0, 0, 0, 0, 0, 0};
        #pragma unroll
        for (int c = 0; c < 4; ++c) acc = wmma4(aq[r][c], bk[ct][c], acc);
        L[r][ct] = acc;
      }

    // ---- logits mean, softmax over f (row across 16 lanes x 2 tiles), attn mean ----
    float* pa = &sA[wave * (Gn * RA)];
    #pragma unroll
    for (int r = 0; r < 2; ++r) {
      ml[r][0] += L[r][0] * 0.125f;
      ml[r][1] += L[r][1] * 0.125f;
      #pragma unroll
      for (int vi = 0; vi < 8; ++vi) {
        float l0 = L[r][0][vi], l1 = L[r][1][vi];
        float m  = red_max16(fmaxf(l0, l1));
        float p0 = __expf(l0 - m);
        float p1 = __expf(l1 - m);
        float inv = __builtin_amdgcn_rcpf(red_sum16(p0 + p1));
        p0 *= inv; p1 *= inv;
        L[r][0][vi] = p0; L[r][1][vi] = p1;
        const int g = r * 16 + vi + 8 * lh;   // D-layout row
        pa[g * RA + lm]      = p0;            // stash attn as [g][f]
        pa[g * RA + 16 + lm] = p1;
      }
      ma[r][0] += L[r][0] * 0.125f;
      ma[r][1] += L[r][1] * 0.125f;
    }
    asm volatile("s_wait_dscnt 0" ::: "memory");  // attn stores visible to our own reads

    // ---- X = attn(32x32) * V(32x16): 2 tiles, K=32 via 8 chained wmma ----
    v2f bv[8];
    #pragma unroll
    for (int c = 0; c < 8; ++c)  // B frag: [K = f][N = d=lm]
      bv[c] = lds2(&sV[(wave * Dn + lm) * RA + c * 4 + 2 * lh]);
    v8f X[2];
    #pragma unroll
    for (int r = 0; r < 2; ++r) {
      v8f acc = {0, 0, 0, 0, 0, 0, 0, 0};
      const float* par = &pa[(r * 16 + lm) * RA];
      #pragma unroll
      for (int c = 0; c < 8; ++c) {
        v2f af = lds2(&par[c * 4 + 2 * lh]);  // A frag: [M = g][K = f]
        acc = wmma4(af, bv[c], acc);
      }
      X[r] = acc;
    }
    __syncthreads();  // all waves done reading sQ before reuse as X stage
    #pragma unroll
    for (int r = 0; r < 2; ++r)
      #pragma unroll
      for (int vi = 0; vi < 8; ++vi) {
        const int g = r * 16 + vi + 8 * lh;
        sQ[(g * Dn + lm) * SC + wave] = X[r][vi];  // [g*16+d][s]
      }
    __syncthreads();
    // coalesced x store: channel = g*128 + h*16 + d, s-contiguous granules
    #pragma unroll
    for (int p = 0; p < 16; ++p) {
      const int row = p * 32 + (tid >> 3);  // g*16 + d, 0..511
      const int si  = tid & 7;
      const int g = row >> 4, d = row & 15;
      outx[((size_t)b * (Gn * HD) + g * HD + h * Dn + d) * Sn + s0 + si] =
          sQ[row * SC + si];
    }
  }

  // ---- head-averaged outputs: stage through LDS, store s-contiguous ----
  __syncthreads();
  #pragma unroll
  for (int r = 0; r < 2; ++r)
    #pragma unroll
    for (int vi = 0; vi < 8; ++vi) {
      const int g = r * 16 + vi + 8 * lh;
      sA[(g * Gn + lm) * SC + wave]      = ma[r][0][vi];
      sA[(g * Gn + 16 + lm) * SC + wave] = ma[r][1][vi];
    }
  __syncthreads();
  #pragma unroll
  for (int p = 0; p < 32; ++p) {
    const int row = p * 32 + (tid >> 3);  // g*32 + f
    const int si  = tid & 7;
    outa[((size_t)b * (Gn * Gn) + row) * Sn + s0 + si] = sA[row * SC + si];
  }
  __syncthreads();
  #pragma unroll
  for (int r = 0; r < 2; ++r)
    #pragma unroll
    for (int vi = 0; vi < 8; ++vi) {
      const int g = r * 16 + vi + 8 * lh;
      sA[(g * Gn + lm) * SC + wave]      = ml[r][0][vi];
      sA[(g * Gn + 16 + lm) * SC + wave] = ml[r][1][vi];
    }
  __syncthreads();
  #pragma unroll
  for (int p = 0; p < 32; ++p) {
    const int row = p * 32 + (tid >> 3);
    const int si  = tid & 7;
    outl[((size_t)b * (Gn * Gn) + row) * Sn + s0 + si] = sA[row * SC + si];
  }
}

extern "C" void kernel_launch(void* const* d_in, const int* in_sizes, int n_in,
                              void* d_out, int out_size, void* d_ws, size_t ws_size,
                              hipStream_t stream) {
  (void)in_sizes; (void)n_in; (void)out_size; (void)d_ws; (void)ws_size;
  const float* q = (const float*)d_in[0];
  const float* k = (const float*)d_in[1];
  const float* v = (const float*)d_in[2];
  float* out = (float*)d_out;
  dim3 grid(Sn / SC, Bn, 1);  // (512, 2) blocks, 256 threads = 8 waves each
  gqa_groupattn_kernel<<<grid, dim3(256, 1, 1), 0, stream>>>(q, k, v, out);
}